// StrictProjectionBlock_26826365731073
// MI455X (gfx1250) — compile-verified
//
#include <hip/hip_runtime.h>
#include <stdint.h>

typedef uint32_t u32;
typedef u32 v4u __attribute__((ext_vector_type(4)));
typedef int  v8i __attribute__((ext_vector_type(8)));
typedef int  v4i __attribute__((ext_vector_type(4)));

#define MATS_PER_BLOCK 256
#define TILE_FLOATS (MATS_PER_BLOCK * 9)

// LDS byte offset of a __shared__ object (generic -> addrspace(3) cast).
static __device__ __forceinline__ u32 lds_byte_off(const void* p) {
  return (u32)(uintptr_t)(__attribute__((address_space(3))) const void*)p;
}

// D# group 0: count=1, lds_addr, 57-bit global addr, type=2 ("image").
static __device__ __forceinline__ v4u tdm_group0(u32 lds_addr, const void* gptr) {
  uint64_t ga = (uint64_t)(uintptr_t)gptr;
  v4u g0;
  g0.x = 1u;                                            // count=1, user mode
  g0.y = lds_addr;                                      // LDS byte address
  g0.z = (u32)ga;                                       // global_addr[31:0]
  g0.w = (u32)((ga >> 32) & 0x01FFFFFFu) | (2u << 30);  // global_addr[56:32] | type=2
  return g0;
}

// D# group 1: 1-D tensor of `nelem` 4-byte elements, tile == tensor.
// Bit layout per ISA 8.4: data_size[17:16]=2 (4B); tensor_dim0 @ [79:48];
// tensor_dim1 @ [111:80] = 1; tile_dim0 @ [127:112]; tile_dim1/2 = 0 (unused);
// tensor_dim0_stride @ [207:160] = nelem.
static __device__ __forceinline__ v8i tdm_group1(u32 nelem) {
  v8i g1;
  g1[0] = (int)(2u << 16);                   // data_size=4B; no multicast/pad/iterate
  g1[1] = (int)((nelem & 0xFFFFu) << 16);    // tensor_dim0[15:0]
  g1[2] = (int)((nelem >> 16) | (1u << 16)); // tensor_dim0[31:16] | tensor_dim1=1
  g1[3] = (int)(nelem << 16);                // tensor_dim1 hi=0 | tile_dim0=nelem
  g1[4] = 0;                                 // tile_dim1=0, tile_dim2=0 (unused)
  g1[5] = (int)nelem;                        // tensor_dim0_stride[31:0]
  g1[6] = 0;                                 // stride0 hi | stride1 lo
  g1[7] = 0;                                 // stride1 hi
  return g1;
}

__global__ __launch_bounds__(MATS_PER_BLOCK)
void strict_projection_tdm_kernel(const float* __restrict__ in,
                                  float* __restrict__ out, int nmat) {
  __shared__ float tile[TILE_FLOATS];
  const int t    = threadIdx.x;
  const int base = (int)blockIdx.x * MATS_PER_BLOCK;
  const int cnt  = min(MATS_PER_BLOCK, nmat - base);
  if (cnt <= 0) return;
  const u32 nel  = (u32)cnt * 9u;

  if (t < 32) {  // wave 0 issues the tile DMA (global -> LDS)
    v4u g0 = tdm_group0(lds_byte_off(tile), in + (size_t)base * 9);
    v8i g1 = tdm_group1(nel);
    v4i z4 = {0, 0, 0, 0};
    v8i z8 = {0, 0, 0, 0, 0, 0, 0, 0};
    __builtin_amdgcn_tensor_load_to_lds(g0, g1, z4, z4, z8, 0);
    __builtin_amdgcn_s_wait_tensorcnt(0);
  }
  __syncthreads();

  if (t < cnt) {
    float q[9];
#pragma unroll
    for (int i = 0; i < 9; ++i) q[i] = tile[t * 9 + i];  // stride-9 words: conflict-free

    // Frobenius normalize
    float s = 0.f;
#pragma unroll
    for (int i = 0; i < 9; ++i) s += q[i] * q[i];
    float inv = 1.0f / sqrtf(s);
#pragma unroll
    for (int i = 0; i < 9; ++i) q[i] *= inv;

    // 4 layers of Q <- Q * (2I - 1.5*N + 0.5*N^2), N = Q^T Q (symmetric)
#pragma unroll
    for (int l = 0; l < 4; ++l) {
      float n00 = q[0]*q[0] + q[3]*q[3] + q[6]*q[6];
      float n01 = q[0]*q[1] + q[3]*q[4] + q[6]*q[7];
      float n02 = q[0]*q[2] + q[3]*q[5] + q[6]*q[8];
      float n11 = q[1]*q[1] + q[4]*q[4] + q[7]*q[7];
      float n12 = q[1]*q[2] + q[4]*q[5] + q[7]*q[8];
      float n22 = q[2]*q[2] + q[5]*q[5] + q[8]*q[8];

      float m00 = n00*n00 + n01*n01 + n02*n02;   // N^2 (symmetric)
      float m01 = n00*n01 + n01*n11 + n02*n12;
      float m02 = n00*n02 + n01*n12 + n02*n22;
      float m11 = n01*n01 + n11*n11 + n12*n12;
      float m12 = n01*n02 + n11*n12 + n12*n22;
      float m22 = n02*n02 + n12*n12 + n22*n22;

      float s00 = 2.0f - 1.5f*n00 + 0.5f*m00;    // S = 2I - 1.5N + 0.5N^2
      float s01 =        -1.5f*n01 + 0.5f*m01;
      float s02 =        -1.5f*n02 + 0.5f*m02;
      float s11 = 2.0f - 1.5f*n11 + 0.5f*m11;
      float s12 =        -1.5f*n12 + 0.5f*m12;
      float s22 = 2.0f - 1.5f*n22 + 0.5f*m22;

#pragma unroll
      for (int r = 0; r < 3; ++r) {              // Q <- Q * S
        float a = q[3*r+0], b = q[3*r+1], c = q[3*r+2];
        q[3*r+0] = a*s00 + b*s01 + c*s02;
        q[3*r+1] = a*s01 + b*s11 + c*s12;
        q[3*r+2] = a*s02 + b*s12 + c*s22;
      }
    }

#pragma unroll
    for (int i = 0; i < 9; ++i) tile[t * 9 + i] = q[i];
  }

  __syncthreads();
  if (t < 32) {  // wave 0 issues the tile DMA (LDS -> global)
    v4u g0 = tdm_group0(lds_byte_off(tile), out + (size_t)base * 9);
    v8i g1 = tdm_group1(nel);
    v4i z4 = {0, 0, 0, 0};
    v8i z8 = {0, 0, 0, 0, 0, 0, 0, 0};
    __builtin_amdgcn_tensor_store_from_lds(g0, g1, z4, z4, z8, 0);
    __builtin_amdgcn_s_wait_tensorcnt(0);
  }
}

extern "C" void kernel_launch(void* const* d_in, const int* in_sizes, int n_in,
                              void* d_out, int out_size, void* d_ws, size_t ws_size,
                              hipStream_t stream) {
  (void)n_in; (void)out_size; (void)d_ws; (void)ws_size;
  const float* in = (const float*)d_in[0];
  float* out = (float*)d_out;
  int nmat = in_sizes[0] / 9;
  int grid = (nmat + MATS_PER_BLOCK - 1) / MATS_PER_BLOCK;
  hipLaunchKernelGGL(strict_projection_tdm_kernel, dim3(grid), dim3(MATS_PER_BLOCK),
                     0, stream, in, out, nmat);
}